// E43ScalarDecayCell_47656957116856
// MI455X (gfx1250) — compile-verified
//
#include <hip/hip_runtime.h>
#include <cstdint>

// Problem constants (reference: T=4096, B=8, D=1024).
#define T_DIM   4096
#define B_DIM   8
#define D_DIM   1024
#define BD      (B_DIM * D_DIM)        // 8192 independent channels
#define NCHUNK  64                     // chunks along T
#define CLEN    (T_DIM / NCHUNK)       // 64 timesteps per chunk
#define NC      1024                   // channels per workgroup
#define CB      (BD / NC)              // 8 channel-blocks
#define TT      8                      // timesteps per TDM tile (32KB tile)
#define NT      (CLEN / TT)            // 8 tiles per chunk
#define THREADS 256

static_assert(CLEN == 64, "lam^CLEN computed by 6 squarings below");

typedef unsigned int v4u __attribute__((ext_vector_type(4)));
typedef int          v8i __attribute__((ext_vector_type(8)));
typedef int          v4i __attribute__((ext_vector_type(4)));
typedef float        v4f __attribute__((ext_vector_type(4)));   // native vec for NT stores

__device__ __forceinline__ float fast_sigmoid(float v)
{
    // v_exp_f32 + v_rcp_f32 (~1 ulp) instead of the IEEE divide expansion.
    return __builtin_amdgcn_rcpf(1.0f + __expf(-v));
}

// ---------------------------------------------------------------------------
// TDM: issue a 2D tensor_load_to_lds of a [TT rows x NC cols] f32 tile.
// Rows are timesteps (stride BD floats in memory), cols are contiguous
// channels. Destination is contiguous LDS at lds_off.
// D# packing per CDNA5 ISA ch.8 (group0 128b, group1 256b; groups 2/3 unused
// for 2D tiles). This toolchain exposes the 6-arg builtin form.
// ---------------------------------------------------------------------------
__device__ __forceinline__ void tdm_load_tile(const float* gsrc, unsigned lds_off)
{
    const unsigned long long ga = (unsigned long long)(uintptr_t)gsrc;

    v4u g0;
    g0[0] = 1u;                                        // count=1, user mode
    g0[1] = lds_off;                                   // lds_addr [63:32]
    g0[2] = (unsigned)(ga & 0xFFFFFFFFull);            // global_addr[31:0]
    g0[3] = (unsigned)((ga >> 32) & 0x1FFFFFFull)      // global_addr[56:32]
          | (2u << 30);                                // type = 2 ("image")

    v8i g1;
    g1[0] = 0x00020000;                                // data_size = 2 (4 bytes)
    g1[1] = (int)((unsigned)(BD & 0xFFFF) << 16);      // tensor_dim0[15:0]
    g1[2] = (int)(((unsigned)(BD >> 16) & 0xFFFFu)     // tensor_dim0[31:16]
          | ((unsigned)(T_DIM & 0xFFFF) << 16));       // tensor_dim1[15:0]
    g1[3] = (int)(((unsigned)(T_DIM >> 16) & 0xFFFFu)  // tensor_dim1[31:16]
          | ((unsigned)NC << 16));                     // tile_dim0 = NC
    g1[4] = TT;                                        // tile_dim1 = TT, tile_dim2 = 0
    g1[5] = BD;                                        // tensor_dim0_stride[31:0]
    g1[6] = 0;                                         // stride hi / dim1_stride lo
    g1[7] = 0;

    const v4i z4 = {0, 0, 0, 0};                       // groups 2/3 unused (2D)
    const v8i z8 = {0, 0, 0, 0, 0, 0, 0, 0};           // trailing group (zero)
    __builtin_amdgcn_tensor_load_to_lds(g0, g1, z4, z4, z8, 0);
}

__device__ __forceinline__ unsigned lds_offset_of(const void* p)
{
    // Generic LDS address: low 32 bits are the workgroup-relative LDS offset.
    return (unsigned)(uintptr_t)p;
}

// ---------------------------------------------------------------------------
// Pass 1 / Pass 3: per-(chunk, channel-block) local scan.
//   FINAL=false: start from h=0, emit only chunk-end partial.
//   FINAL=true : start from carry[chunk], emit out[t] and h[t+1].
// x tiles streamed through LDS by the Tensor Data Mover (wave 0 issues,
// TENSORcnt-gated, double-buffered, barrier hand-off to the other waves).
// ---------------------------------------------------------------------------
template <bool FINAL>
__global__ __launch_bounds__(THREADS)
void scan_chunk_kernel(const float* __restrict__ x,
                       const float* __restrict__ log_lambda,
                       const float* __restrict__ bias,
                       const float* __restrict__ carry,    // FINAL only
                       float*       __restrict__ partial,  // !FINAL only
                       float*       __restrict__ out,      // FINAL only
                       float*       __restrict__ hout)     // FINAL only (h base)
{
    __shared__ float smem[2][TT * NC];                  // 2 x 32KB double buffer

    const int chunk = blockIdx.x / CB;
    const int cb    = blockIdx.x % CB;
    const int c0    = cb * NC;
    const int tid   = threadIdx.x;
    const int c     = c0 + tid * 4;                     // 4 channels per thread

    const float lam = fast_sigmoid(log_lambda[0]);
    const v4f b4 = *(const v4f*)(bias + (c & (D_DIM - 1)));

    v4f h4;
    if (FINAL) {
        h4 = *(const v4f*)(carry + (size_t)chunk * BD + c);
    } else {
        h4 = (v4f){0.f, 0.f, 0.f, 0.f};
    }

    const float* gx = x + (size_t)(chunk * CLEN) * BD + c0;
    const bool issuer = (tid < 32);                     // exactly wave 0

    if (issuer) {
        tdm_load_tile(gx + (size_t)0 * TT * BD, lds_offset_of(&smem[0][0]));
        tdm_load_tile(gx + (size_t)1 * TT * BD, lds_offset_of(&smem[1][0]));
    }

    for (int it = 0; it < NT; ++it) {
        if (issuer) {
            // In-order TDM per wave: tensorcnt<=1 => tile `it` has landed.
            if (it == NT - 1) __builtin_amdgcn_s_wait_tensorcnt(0);
            else              __builtin_amdgcn_s_wait_tensorcnt(1);
        }
        __syncthreads();                                // tile `it` visible to all

        const float* buf = &smem[it & 1][0];
        #pragma unroll
        for (int tt = 0; tt < TT; ++tt) {
            const v4f x4 = *(const v4f*)(buf + tt * NC + tid * 4);
            h4.x = lam * (x4.x + h4.x) + b4.x;
            h4.y = lam * (x4.y + h4.y) + b4.y;
            h4.z = lam * (x4.z + h4.z) + b4.z;
            h4.w = lam * (x4.w + h4.w) + b4.w;
            if (FINAL) {
                const int t = chunk * CLEN + it * TT + tt;
                v4f o4;
                o4.x = h4.x * h4.x * fast_sigmoid(h4.x);
                o4.y = h4.y * h4.y * fast_sigmoid(h4.y);
                o4.z = h4.z * h4.z * fast_sigmoid(h4.z);
                o4.w = h4.w * h4.w * fast_sigmoid(h4.w);
                // Write-once streams (268MB total > 192MB L2): non-temporal.
                __builtin_nontemporal_store(o4, (v4f*)(out  + (size_t)t       * BD + c));
                __builtin_nontemporal_store(h4, (v4f*)(hout + (size_t)(t + 1) * BD + c));
            }
        }
        __syncthreads();                                // everyone done with buf
        if (issuer && (it + 2) < NT) {
            tdm_load_tile(gx + (size_t)(it + 2) * TT * BD,
                          lds_offset_of(&smem[it & 1][0]));
        }
    }

    if (!FINAL) {
        *(v4f*)(partial + (size_t)chunk * BD + c) = h4;
    }
}

// ---------------------------------------------------------------------------
// Pass 2: serial scan over the 64 chunk partials per channel (tiny: 2MB in,
// 2MB out). carry[j] = state entering chunk j; H_{j+1} = lam^64 * H_j + p_j.
// Also writes h[0] = h0.
// ---------------------------------------------------------------------------
__global__ __launch_bounds__(256)
void chunk_scan_kernel(const float* __restrict__ h0,
                       const float* __restrict__ log_lambda,
                       const float* __restrict__ partial,
                       float*       __restrict__ carry,
                       float*       __restrict__ h_row0)
{
    const int c = blockIdx.x * blockDim.x + threadIdx.x;   // 0..BD-1
    const float lam = fast_sigmoid(log_lambda[0]);
    const float l2 = lam * lam, l4 = l2 * l2, l8 = l4 * l4;
    const float l16 = l8 * l8, l32 = l16 * l16, lamL = l32 * l32;   // lam^64

    float H = h0[c];
    h_row0[c] = H;                                         // h[0] = h0
    #pragma unroll 8
    for (int j = 0; j < NCHUNK; ++j) {
        carry[(size_t)j * BD + c] = H;
        H = lamL * H + partial[(size_t)j * BD + c];
    }
}

// ---------------------------------------------------------------------------
// Entry point. d_in: x[T,B,D] f32, h0[B,D] f32, log_lambda[1] f32, b[D] f32.
// d_out: out[T,B,D] followed by h[T+1,B,D], f32.
// d_ws: needs 4MB (partials 2MB + carries 2MB).
// ---------------------------------------------------------------------------
extern "C" void kernel_launch(void* const* d_in, const int* in_sizes, int n_in,
                              void* d_out, int out_size, void* d_ws, size_t ws_size,
                              hipStream_t stream)
{
    (void)in_sizes; (void)n_in; (void)out_size; (void)ws_size;

    const float* x   = (const float*)d_in[0];
    const float* h0  = (const float*)d_in[1];
    const float* ll  = (const float*)d_in[2];
    const float* b   = (const float*)d_in[3];

    float* out     = (float*)d_out;                     // [T, B, D]
    float* h       = out + (size_t)T_DIM * BD;          // [T+1, B, D]
    float* partial = (float*)d_ws;                      // [NCHUNK, BD]
    float* carry   = partial + (size_t)NCHUNK * BD;     // [NCHUNK, BD]

    const dim3 grid(NCHUNK * CB);                       // 512 blocks

    scan_chunk_kernel<false><<<grid, THREADS, 0, stream>>>(
        x, ll, b, nullptr, partial, nullptr, nullptr);

    chunk_scan_kernel<<<BD / 256, 256, 0, stream>>>(h0, ll, partial, carry, h);

    scan_chunk_kernel<true><<<grid, THREADS, 0, stream>>>(
        x, ll, b, carry, nullptr, out, h);
}